// FSI_Initial_12352325944215
// MI455X (gfx1250) — compile-verified
//
#include <hip/hip_runtime.h>
#include <stdint.h>

// Problem constants (match reference: B,H,W,C = 8,512,512,16 ; P=10)
#define HH 512
#define WW 512
#define CC 16
#define BB 8
#define HW (HH * WW)          // 262144
#define MM (HW * CC)          // 4194304
#define KRANK 3774873u        // M - ceil(M/10) = 4194304 - 419431

// Workspace layout (uint32 indices)
#define HIST1   0
#define HIST2   (HIST1 + BB * 2048)   // 16384
#define HIST3   (HIST2 + BB * 2048)   // 32768
#define SEL1P   (HIST3 + BB * 1024)   // 40960
#define SEL1K   (SEL1P + BB)
#define SEL2P   (SEL1K + BB)
#define SEL2K   (SEL2P + BB)
#define THRBITS (SEL2K + BB)
#define WS_U32  (THRBITS + BB)        // 41000 u32 = 164 KB

// Monotonic key: order on keys == order on floats (sign-flip trick)
__device__ __forceinline__ uint32_t fkey(float f) {
  uint32_t u = __float_as_uint(f);
  uint32_t m = ((uint32_t)((int32_t)u >> 31)) | 0x80000000u;
  return u ^ m;
}

// CDNA5 async copy: global -> LDS, 16B per lane (GVS mode: sgpr base + vgpr u32 offset)
__device__ __forceinline__ void async_ld_b128(uint32_t lds_byte_off,
                                              const float* gbase,
                                              uint32_t byte_off) {
  asm volatile("global_load_async_to_lds_b128 %0, %1, %2"
               :
               : "v"(lds_byte_off), "v"(byte_off), "s"(gbase)
               : "memory");
}
__device__ __forceinline__ void wait_async0() {
  asm volatile("s_wait_asynccnt 0" ::: "memory");
}

__global__ void zero_ws_kernel(uint32_t* __restrict__ ws) {
  int i = blockIdx.x * 256 + threadIdx.x;
  if (i < WS_U32) ws[i] = 0u;
}

// Radix-select histogram pass. LEVEL 0: bits[31:21]; LEVEL 1: bits[20:10]
// filtered by 11-bit prefix; LEVEL 2: bits[9:0] filtered by 22-bit prefix.
// Each block handles a 4096-float chunk, staged through LDS via async loads.
template <int LEVEL>
__global__ __launch_bounds__(256) void hist_kernel(const float* __restrict__ Y,
                                                   uint32_t* __restrict__ ws) {
  constexpr int NB = (LEVEL == 2) ? 1024 : 2048;
  __shared__ uint32_t lh[NB];
  __shared__ float stage[4096];   // 16 KB staging tile

  const int t = threadIdx.x;
  const int b = blockIdx.y;

  for (int i = t; i < NB; i += 256) lh[i] = 0u;

  uint32_t pfx = 0u;
  if (LEVEL == 1) pfx = ws[SEL1P + b];
  if (LEVEL == 2) pfx = ws[SEL2P + b];
  __syncthreads();

  const float* gbase = Y + (size_t)b * MM + (size_t)blockIdx.x * 4096;

  // Issue 4 async b128 loads per thread (each wave writes/reads only its own
  // staging slots, so no block barrier is needed around the staging).
#pragma unroll
  for (int i = 0; i < 4; ++i) {
    const int fidx = i * 1024 + t * 4;
    uint32_t lds_off = (uint32_t)(uintptr_t)(&stage[fidx]); // low 32 bits = LDS addr
    async_ld_b128(lds_off, gbase, (uint32_t)(fidx * 4));
  }
  wait_async0();

#pragma unroll
  for (int i = 0; i < 4; ++i) {
    const int fidx = i * 1024 + t * 4;
    float4 v = *reinterpret_cast<const float4*>(&stage[fidx]);
    float vals[4] = {v.x, v.y, v.z, v.w};
#pragma unroll
    for (int j = 0; j < 4; ++j) {
      uint32_t key = fkey(vals[j]);
      if (LEVEL == 0) {
        atomicAdd(&lh[key >> 21], 1u);
      } else if (LEVEL == 1) {
        if ((key & 0xFFE00000u) == pfx) atomicAdd(&lh[(key >> 10) & 0x7FFu], 1u);
      } else {
        if ((key & 0xFFFFFC00u) == pfx) atomicAdd(&lh[key & 0x3FFu], 1u);
      }
    }
  }
  __syncthreads();

  uint32_t* gh = ws + (LEVEL == 0 ? HIST1 : (LEVEL == 1 ? HIST2 : HIST3)) + (size_t)b * NB;
  for (int i = t; i < NB; i += 256) {
    uint32_t c = lh[i];
    if (c) atomicAdd(&gh[i], c);
  }
}

// Find which bin holds rank k; record refined prefix / residual rank.
// One block per batch.
template <int LEVEL>
__global__ __launch_bounds__(256) void scan_kernel(uint32_t* __restrict__ ws) {
  constexpr int NB  = (LEVEL == 2) ? 1024 : 2048;
  constexpr int PER = NB / 256;
  __shared__ uint32_t part[256];

  const int t = threadIdx.x;
  const int b = blockIdx.x;
  const uint32_t* gh =
      ws + (LEVEL == 0 ? HIST1 : (LEVEL == 1 ? HIST2 : HIST3)) + (size_t)b * NB;

  uint32_t k = (LEVEL == 0) ? KRANK : ws[(LEVEL == 1 ? SEL1K : SEL2K) + b];

  uint32_t mycnt[PER];
  uint32_t s = 0u;
#pragma unroll
  for (int j = 0; j < PER; ++j) { mycnt[j] = gh[t * PER + j]; s += mycnt[j]; }
  part[t] = s;
  __syncthreads();
  if (t == 0) {
    uint32_t run = 0u;
    for (int i = 0; i < 256; ++i) { uint32_t c = part[i]; part[i] = run; run += c; }
  }
  __syncthreads();

  uint32_t cum = part[t];
#pragma unroll
  for (int j = 0; j < PER; ++j) {
    uint32_t c = mycnt[j];
    if (k >= cum && k < cum + c) {   // exactly one thread/bin satisfies this
      uint32_t bin  = (uint32_t)(t * PER + j);
      uint32_t krem = k - cum;
      if (LEVEL == 0) {
        ws[SEL1P + b] = bin << 21;
        ws[SEL1K + b] = krem;
      } else if (LEVEL == 1) {
        ws[SEL2P + b] = ws[SEL1P + b] | (bin << 10);
        ws[SEL2K + b] = krem;
      } else {
        uint32_t key = ws[SEL2P + b] | bin;    // exact bit pattern of k-th value
        uint32_t u = (key & 0x80000000u) ? (key & 0x7FFFFFFFu) : ~key;
        ws[THRBITS + b] = u;
      }
    }
    cum += c;
  }
}

// out[b,c,h,w] = (Y[b,h,w,c] >= thresh[b]) ? 1 : 0   (write-once -> NT stores)
__global__ __launch_bounds__(256) void output_kernel(const float* __restrict__ Y,
                                                     float* __restrict__ out,
                                                     const uint32_t* __restrict__ ws) {
  const int t = threadIdx.x;
  const int b = blockIdx.y;
  const int idx = blockIdx.x * 256 + t;     // hw index, 0..HW-1
  const float th = __uint_as_float(ws[THRBITS + b]);

  const float4* src =
      reinterpret_cast<const float4*>(Y + ((size_t)b * HW + (size_t)idx) * CC);
  float vals[CC];
#pragma unroll
  for (int i = 0; i < 4; ++i) {
    float4 v = src[i];
    vals[i * 4 + 0] = v.x; vals[i * 4 + 1] = v.y;
    vals[i * 4 + 2] = v.z; vals[i * 4 + 3] = v.w;
  }
  float* obase = out + (size_t)b * CC * HW + (size_t)idx;
#pragma unroll
  for (int c = 0; c < CC; ++c) {
    float r = (vals[c] >= th) ? 1.0f : 0.0f;
    __builtin_nontemporal_store(r, obase + (size_t)c * HW);
  }
}

extern "C" void kernel_launch(void* const* d_in, const int* in_sizes, int n_in,
                              void* d_out, int out_size, void* d_ws, size_t ws_size,
                              hipStream_t stream) {
  (void)in_sizes; (void)n_in; (void)out_size; (void)ws_size;
  const float* Y  = (const float*)d_in[0];
  const float* Z0 = (const float*)d_in[1];     // complex64 as interleaved floats
  float* out      = (float*)d_out;
  uint32_t* ws    = (uint32_t*)d_ws;

  zero_ws_kernel<<<(WS_U32 + 255) / 256, 256, 0, stream>>>(ws);

  dim3 hgrid(MM / 4096, BB);
  hist_kernel<0><<<hgrid, 256, 0, stream>>>(Y, ws);
  scan_kernel<0><<<BB, 256, 0, stream>>>(ws);
  hist_kernel<1><<<hgrid, 256, 0, stream>>>(Y, ws);
  scan_kernel<1><<<BB, 256, 0, stream>>>(ws);
  hist_kernel<2><<<hgrid, 256, 0, stream>>>(Y, ws);
  scan_kernel<2><<<BB, 256, 0, stream>>>(ws);

  dim3 ogrid(HW / 256, BB);
  output_kernel<<<ogrid, 256, 0, stream>>>(Y, out, ws);

  // Z0 pass-through: (1,1,H,W) complex64 = 2*HW floats appended after Ytr
  hipMemcpyAsync(out + (size_t)BB * CC * HW, Z0,
                 (size_t)2 * HW * sizeof(float),
                 hipMemcpyDeviceToDevice, stream);
}